// GTN_39187281608743
// MI455X (gfx1250) — compile-verified
//
#include <hip/hip_runtime.h>

#define N_USERS  100000
#define N_ITEMS  50000
#define N_NODES  (N_USERS + N_ITEMS)
#define NNZ      4000000
#define BATCH    16384
#define DIM      64
#define ALPHA    0.1f
#define K_LAYERS 3

typedef __attribute__((ext_vector_type(2))) float v2f;
typedef __attribute__((ext_vector_type(8))) float v8f;

// ---------------------------------------------------------------------------
// SpMM scatter: one wave per edge. 32 lanes * float2 = the full 64-dim row.
// Gather x[col] (coalesced 256B), scale by w, scatter-add into agg[row] with
// hardware f32 atomics (non-returning -> resolved at L2; agg fits in 192MB L2).
// ---------------------------------------------------------------------------
__global__ __launch_bounds__(256) void GTN_spmm_scatter(
    const float* __restrict__ x, const float* __restrict__ vals,
    const int* __restrict__ row, const int* __restrict__ col,
    float* __restrict__ agg)
{
    unsigned gid  = blockIdx.x * blockDim.x + threadIdx.x;
    unsigned e    = gid >> 5;
    unsigned lane = gid & 31;
    if (e >= NNZ) return;

    int   r = row[e];
    int   c = col[e];
    float w = vals[e];

    v2f xv = ((const v2f*)(x + (size_t)c * DIM))[lane];
    float* dst = agg + (size_t)r * DIM + lane * 2;
    unsafeAtomicAdd(dst + 0, w * xv.x);   // global_atomic_add_f32
    unsafeAtomicAdd(dst + 1, w * xv.y);
}

// ---------------------------------------------------------------------------
// x = ALPHA * x0 + (1-ALPHA) * agg   (x0 read straight from the input embs)
// ---------------------------------------------------------------------------
__global__ __launch_bounds__(256) void GTN_combine(
    const float* __restrict__ ue, const float* __restrict__ ie,
    const float* __restrict__ agg, float* __restrict__ x)
{
    size_t idx = (size_t)blockIdx.x * blockDim.x + threadIdx.x;
    const size_t NU  = (size_t)N_USERS * DIM;
    const size_t TOT = (size_t)N_NODES * DIM;
    if (idx >= TOT) return;
    float x0 = (idx < NU) ? ue[idx] : ie[idx - NU];
    x[idx] = ALPHA * x0 + (1.0f - ALPHA) * agg[idx];
}

// ---------------------------------------------------------------------------
// Scoring: 16 (user,item) pairs per wave. D = U * I^T accumulated over K=64
// via 16 chained V_WMMA_F32_16X16X4_F32; gamma = diag(D). Exact f32 math.
//
// A 16x4 f32 layout (ISA 7.12.2): lane l holds A[l&15][2*(l>>4) + {0,1}]
// B 4x16 f32 layout (mirror):     lane l holds B[2*(l>>4)+{0,1}][l&15]
//    with B[k][n] = I[n][k]  (so B rows are item features along K)
// C/D 16x16 f32: VGPR v, lanes 0-15 -> D[v][l]; lanes 16-31 -> D[v+8][l-16]
//    diag b<8  at lane b,    c[b]
//    diag b>=8 at lane b+16, c[b-8]
// ---------------------------------------------------------------------------
__global__ __launch_bounds__(64) void GTN_score_wmma(
    const float* __restrict__ x, const int* __restrict__ users,
    const int* __restrict__ items, float* __restrict__ out)
{
    unsigned lane = threadIdx.x & 31;
    unsigned tile = blockIdx.x * 2 + (threadIdx.x >> 5);   // 2 waves / block

    unsigned m  = lane & 15;          // row of A / col of B this lane feeds
    unsigned kh = (lane >> 4) << 1;   // K sub-offset: 0 or 2

    int un = users[tile * 16 + m];
    int it = items[tile * 16 + m];
    const float* urow = x + (size_t)un * DIM;
    const float* irow = x + ((size_t)N_USERS + (size_t)it) * DIM;

    v8f c = {0.f, 0.f, 0.f, 0.f, 0.f, 0.f, 0.f, 0.f};
#pragma unroll
    for (int kc = 0; kc < DIM / 4; ++kc) {
        v2f a = *(const v2f*)(urow + kc * 4 + kh);
        v2f b = *(const v2f*)(irow + kc * 4 + kh);
        // 8 args: (neg_a, A, neg_b, B, c_mod, C, reuse_a, reuse_b)
        c = __builtin_amdgcn_wmma_f32_16x16x4_f32(
                false, a, false, b, (short)0, c, false, false);
    }

    // pull the diagonal out
    int b_idx = -1, v = 0;
    if (lane < 8)        { b_idx = (int)lane;      v = (int)lane;      }
    else if (lane >= 24) { b_idx = (int)lane - 16; v = (int)lane - 24; }
    if (b_idx >= 0) {
        float val = 0.f;
#pragma unroll
        for (int k = 0; k < 8; ++k) if (v == k) val = c[k];
        out[tile * 16 + b_idx] = val;
    }
}

extern "C" void kernel_launch(void* const* d_in, const int* in_sizes, int n_in,
                              void* d_out, int out_size, void* d_ws, size_t ws_size,
                              hipStream_t stream)
{
    const float* user_emb  = (const float*)d_in[0];
    const float* item_emb  = (const float*)d_in[1];
    const float* edge_vals = (const float*)d_in[2];
    const int*   row       = (const int*)d_in[3];
    const int*   col       = (const int*)d_in[4];
    const int*   users     = (const int*)d_in[5];
    const int*   items     = (const int*)d_in[6];
    float*       out       = (float*)d_out;

    float* x   = (float*)d_ws;                         // 150000*64 f32 = 38.4 MB
    float* agg = x + (size_t)N_NODES * DIM;            // another 38.4 MB

    const size_t ubytes = (size_t)N_USERS * DIM * sizeof(float);
    const size_t ibytes = (size_t)N_ITEMS * DIM * sizeof(float);
    const size_t nbytes = (size_t)N_NODES * DIM * sizeof(float);

    // x = x0 = concat(user_emb, item_emb)
    hipMemcpyAsync(x, user_emb, ubytes, hipMemcpyDeviceToDevice, stream);
    hipMemcpyAsync(x + (size_t)N_USERS * DIM, item_emb, ibytes,
                   hipMemcpyDeviceToDevice, stream);

    const unsigned spmm_blocks = (unsigned)(((size_t)NNZ * 32 + 255) / 256);
    const unsigned comb_blocks = (unsigned)(((size_t)N_NODES * DIM + 255) / 256);

    for (int layer = 0; layer < K_LAYERS; ++layer) {
        hipMemsetAsync(agg, 0, nbytes, stream);
        GTN_spmm_scatter<<<spmm_blocks, 256, 0, stream>>>(x, edge_vals, row, col, agg);
        GTN_combine<<<comb_blocks, 256, 0, stream>>>(user_emb, item_emb, agg, x);
    }

    GTN_score_wmma<<<BATCH / 16 / 2, 64, 0, stream>>>(x, users, items, out);
}